// VectorQuantizer2DHS_53412213293158
// MI455X (gfx1250) — compile-verified
//
#include <hip/hip_runtime.h>
#include <hip/hip_bf16.h>

typedef __attribute__((ext_vector_type(2))) float v2f;
typedef __attribute__((ext_vector_type(8))) float v8f;

#define NE      1024
#define EDIM    256
#define BATCH   64
#define TOKENS  16384            // 64*256
#define ZQ_N    4194304          // TOKENS*EDIM
#define BETA    0.25f

// d_out float offsets (tuple concatenated flat, return order)
#define OUT_ZQ    0
#define OUT_LOSS  4194304
#define OUT_SAMP  4194305        // 64*1024 = 65536 entries
#define OUT_IDX   4259841        // 16384 entries (indices as float)
#define OUT_CBVAR 4276225
#define OUT_MIND  4276226
#define OUT_HSW   4276227
#define OUT_MEANR 4276228

// workspace float offsets
#define WS_SQ    0   // sum (z_q - z)^2
#define WS_HSW   1   // sum (r - norm)^2
#define WS_RSUM  2   // sum r
#define WS_VAR   3   // mean row variance (accumulated /NE)
#define WS_MIND  4   // mean 2nd-smallest (accumulated /NE)
#define WS_NORMS 8
#define WS_ESQ   (8 + NE)
#define WS_IDX   (8 + 2*NE)      // int[TOKENS]

__device__ __forceinline__ float blockReduceSumF(float v, float* sh) {
  int t = threadIdx.x;
  sh[t] = v; __syncthreads();
  for (int s = blockDim.x >> 1; s > 0; s >>= 1) {
    if (t < s) sh[t] += sh[t + s];
    __syncthreads();
  }
  float r = sh[0]; __syncthreads();
  return r;
}

// ---------------- K0: zero accumulators + sampled_idx region ----------------
__global__ __launch_bounds__(256) void k0_init(float* out, float* ws) {
  int i = blockIdx.x * 256 + threadIdx.x;
  if (i < BATCH * NE) out[OUT_SAMP + i] = 0.0f;
  if (i < 8) ws[i] = 0.0f;
}

// ---------------- K1: per-code norms and squared norms ----------------
__global__ __launch_bounds__(256) void k1_norms(const float* __restrict__ cb, float* ws) {
  __shared__ float sh[256];
  int row = blockIdx.x;
  float v = cb[row * EDIM + threadIdx.x];
  float s = blockReduceSumF(v * v, sh);
  if (threadIdx.x == 0) {
    ws[WS_NORMS + row] = sqrtf(s);
    ws[WS_ESQ + row]   = s;
  }
}

// ---------------- K1b: hsw and mean(r) partial sums ----------------
__global__ __launch_bounds__(256) void k1b_hsw(const float* __restrict__ r, float* ws) {
  __shared__ float sh[256];
  int i = blockIdx.x * 256 + threadIdx.x;
  float rv = r[i];
  float d  = rv - ws[WS_NORMS + i];
  float s1 = blockReduceSumF(d * d, sh);
  float s2 = blockReduceSumF(rv, sh);
  if (threadIdx.x == 0) {
    atomicAdd(&ws[WS_HSW], s1);
    atomicAdd(&ws[WS_RSUM], s2);
  }
}

// ---------------- K2: gram row stats (variance + 2nd smallest angle) -------
__global__ __launch_bounds__(256) void k2_gram(const float* __restrict__ cb, float* ws) {
  __shared__ float srow[EDIM];
  __shared__ double shd[256];
  __shared__ float sm1[256];
  __shared__ float sm2[256];
  int i = blockIdx.x;
  int t = threadIdx.x;
  srow[t] = cb[i * EDIM + t];
  __syncthreads();

  float ni = ws[WS_NORMS + i];
  double sum = 0.0, sumsq = 0.0;
  float m1 = 1e30f, m2 = 1e30f;
  const float4* srow4 = (const float4*)srow;
  for (int jj = 0; jj < 4; ++jj) {
    int j = t + jj * 256;
    const float4* rj = (const float4*)(cb + (size_t)j * EDIM);
    float dot = 0.0f;
    #pragma unroll 8
    for (int k = 0; k < EDIM / 4; ++k) {
      float4 a = srow4[k];
      float4 b = rj[k];
      dot += a.x * b.x + a.y * b.y + a.z * b.z + a.w * b.w;
    }
    float e = dot / (ni * ws[WS_NORMS + j]);
    e = fminf(fmaxf(e, -0.99999f), 0.99999f);
    float dv = acosf(e);
    sum += (double)dv;
    sumsq += (double)dv * (double)dv;
    if (dv < m1) { m2 = m1; m1 = dv; }
    else if (dv < m2) { m2 = dv; }
  }

  // reduce sum (double)
  shd[t] = sum; __syncthreads();
  for (int s = 128; s > 0; s >>= 1) { if (t < s) shd[t] += shd[t + s]; __syncthreads(); }
  double rsum = shd[0]; __syncthreads();
  shd[t] = sumsq; __syncthreads();
  for (int s = 128; s > 0; s >>= 1) { if (t < s) shd[t] += shd[t + s]; __syncthreads(); }
  double rsumsq = shd[0]; __syncthreads();

  // reduce two-smallest
  sm1[t] = m1; sm2[t] = m2; __syncthreads();
  for (int s = 128; s > 0; s >>= 1) {
    if (t < s) {
      float a1 = sm1[t], a2 = sm2[t], b1 = sm1[t + s], b2 = sm2[t + s];
      float n1 = fminf(a1, b1);
      float n2 = fminf(fmaxf(a1, b1), fminf(a2, b2));
      sm1[t] = n1; sm2[t] = n2;
    }
    __syncthreads();
  }
  if (t == 0) {
    double var = (rsumsq - rsum * rsum / (double)NE) / (double)(NE - 1);
    atomicAdd(&ws[WS_VAR],  (float)(var / (double)NE));
    atomicAdd(&ws[WS_MIND], sm2[0] * (1.0f / (float)NE));
  }
}

// ---------------- K3: WMMA f32 GEMM + argmin ----------------
// 2 waves/block, each wave owns a 16-token tile cached in LDS (32 KB total).
// 4 code tiles (64 codes) per outer iteration -> 4 independent WMMA chains,
// A fragment reused 4x per ds load.
__global__ __launch_bounds__(64) void k3_argmin(const float* __restrict__ z,
                                                const float* __restrict__ cb,
                                                const float* __restrict__ ws,
                                                int* __restrict__ out_idx) {
  __shared__ float sA[2 * 16 * EDIM];   // 32 KB
  const int wave = threadIdx.x >> 5;
  const int lane = threadIdx.x & 31;
  const int m0   = (blockIdx.x * 2 + wave) * 16;
  float* myA = sA + wave * 16 * EDIM;

  // stage 16x256 token tile into LDS (float4)
  {
    const float4* src = (const float4*)(z + (size_t)m0 * EDIM);
    float4* dst = (float4*)myA;
    for (int i = lane; i < 16 * EDIM / 4; i += 32) dst[i] = src[i];
  }
  __syncthreads();

  const int half = lane >> 4;       // 0: K pair {0,1}, 1: K pair {2,3}
  const int l16  = lane & 15;       // M row (A) / N col (B,D)
  const float* arow = myA + l16 * EDIM;
  const float* esq  = ws + WS_ESQ;

  float minv[8]; int mini[8];
  #pragma unroll
  for (int v = 0; v < 8; ++v) { minv[v] = 3.0e38f; mini[v] = 0; }

  for (int n0 = 0; n0 < NE; n0 += 64) {
    const float* brow0 = cb + (size_t)(n0 +  0 + l16) * EDIM;
    const float* brow1 = cb + (size_t)(n0 + 16 + l16) * EDIM;
    const float* brow2 = cb + (size_t)(n0 + 32 + l16) * EDIM;
    const float* brow3 = cb + (size_t)(n0 + 48 + l16) * EDIM;
    v8f acc0 = {0.f,0.f,0.f,0.f,0.f,0.f,0.f,0.f};
    v8f acc1 = {0.f,0.f,0.f,0.f,0.f,0.f,0.f,0.f};
    v8f acc2 = {0.f,0.f,0.f,0.f,0.f,0.f,0.f,0.f};
    v8f acc3 = {0.f,0.f,0.f,0.f,0.f,0.f,0.f,0.f};
    #pragma unroll 4
    for (int k = 0; k < EDIM; k += 4) {
      const int kk = k + 2 * half;
      v2f a, b0, b1, b2, b3;
      a.x  = arow[kk];  a.y  = arow[kk + 1];
      b0.x = brow0[kk]; b0.y = brow0[kk + 1];
      b1.x = brow1[kk]; b1.y = brow1[kk + 1];
      b2.x = brow2[kk]; b2.y = brow2[kk + 1];
      b3.x = brow3[kk]; b3.y = brow3[kk + 1];
      acc0 = __builtin_amdgcn_wmma_f32_16x16x4_f32(false, a, false, b0,
                                                   (short)0, acc0, false, false);
      acc1 = __builtin_amdgcn_wmma_f32_16x16x4_f32(false, a, false, b1,
                                                   (short)0, acc1, false, false);
      acc2 = __builtin_amdgcn_wmma_f32_16x16x4_f32(false, a, false, b2,
                                                   (short)0, acc2, false, false);
      acc3 = __builtin_amdgcn_wmma_f32_16x16x4_f32(false, a, false, b3,
                                                   (short)0, acc3, false, false);
    }
    // dist(m, n) = ||e_n||^2 - 2 * dot ; lane holds N, M = v + 8*half
    #pragma unroll
    for (int t = 0; t < 4; ++t) {
      const v8f acc = (t == 0) ? acc0 : (t == 1) ? acc1 : (t == 2) ? acc2 : acc3;
      const int idx = n0 + 16 * t + l16;
      const float es = esq[idx];
      #pragma unroll
      for (int v = 0; v < 8; ++v) {
        float dist = es - 2.0f * acc[v];
        if (dist < minv[v]) { minv[v] = dist; mini[v] = idx; }  // strict < keeps lowest idx
      }
    }
  }

  // cross-lane argmin within each 16-lane group (first-occurrence tie-break)
  #pragma unroll
  for (int v = 0; v < 8; ++v) {
    float mv = minv[v]; int mi = mini[v];
    #pragma unroll
    for (int off = 8; off >= 1; off >>= 1) {
      float ov = __shfl_xor(mv, off, 16);
      int   oi = __shfl_xor(mi, off, 16);
      if (ov < mv || (ov == mv && oi < mi)) { mv = ov; mi = oi; }
    }
    if (l16 == 0) out_idx[m0 + v + half * 8] = mi;
  }
}

// ---------------- K4: gather z_q, outputs, loss partial ----------------
__global__ __launch_bounds__(256) void k4_gather(const float* __restrict__ z,
                                                 const float* __restrict__ cb,
                                                 const int* __restrict__ idx,
                                                 float* __restrict__ out,
                                                 float* __restrict__ ws) {
  __shared__ float sh[256];
  int tkn = blockIdx.x;
  int t = threadIdx.x;
  int id = idx[tkn];
  float zv = z[(size_t)tkn * EDIM + t];
  float q  = cb[(size_t)id * EDIM + t];
  float diff = q - zv;
  out[OUT_ZQ + (size_t)tkn * EDIM + t] = zv + diff;   // z + sg(z_q - z)
  float s = blockReduceSumF(diff * diff, sh);
  if (t == 0) {
    atomicAdd(&ws[WS_SQ], s);
    out[OUT_SAMP + id] = 1.0f;          // faithful flat scatter (row 0 only)
    out[OUT_IDX + tkn] = (float)id;
  }
}

// ---------------- K5: finalize scalars ----------------
__global__ void k5_final(float* out, const float* ws) {
  float hsw   = ws[WS_HSW] * (1.0f / (float)NE);
  float meanr = ws[WS_RSUM] * (1.0f / (float)NE);
  float cbvar = ws[WS_VAR];
  float mind  = ws[WS_MIND];
  float mse   = ws[WS_SQ] * (1.0f / (float)ZQ_N);
  out[OUT_LOSS]  = (1.0f + BETA) * mse + hsw + (cbvar - mind);
  out[OUT_CBVAR] = cbvar;
  out[OUT_MIND]  = mind;
  out[OUT_HSW]   = hsw;
  out[OUT_MEANR] = meanr;
}

extern "C" void kernel_launch(void* const* d_in, const int* in_sizes, int n_in,
                              void* d_out, int out_size, void* d_ws, size_t ws_size,
                              hipStream_t stream) {
  const float* z  = (const float*)d_in[0];
  const float* cb = (const float*)d_in[1];
  const float* r  = (const float*)d_in[2];
  float* out = (float*)d_out;
  float* ws  = (float*)d_ws;
  int*   idx = (int*)(ws + WS_IDX);

  k0_init  <<<(BATCH * NE + 255) / 256, 256, 0, stream>>>(out, ws);
  k1_norms <<<NE, 256, 0, stream>>>(cb, ws);
  k1b_hsw  <<<NE / 256, 256, 0, stream>>>(r, ws);
  k2_gram  <<<NE, 256, 0, stream>>>(cb, ws);
  k3_argmin<<<TOKENS / 32, 64, 0, stream>>>(z, cb, ws, idx);
  k4_gather<<<TOKENS, 256, 0, stream>>>(z, cb, idx, out, ws);
  k5_final <<<1, 1, 0, stream>>>(out, ws);
}